// CRFLayer_20160576487634
// MI455X (gfx1250) — compile-verified
//
#include <hip/hip_runtime.h>
#include <math.h>

typedef float v2f __attribute__((ext_vector_type(2)));
typedef float v8f __attribute__((ext_vector_type(8)));

#define NB 256
#define NS 256
#define ND 768
#define NT 21
#define NROWS (NB * NS)

// ---------------------------------------------------------------------------
// Phase 1: emissions = features @ fc_w + fc_b   (65536x768 @ 768x21, fp32)
// HBM-bound (201 MB feature read @ 23.3 TB/s ~ 8.6 us floor).
// LDS-staged tiles, v_wmma_f32_16x16x4_f32 per ISA 7.12.2 layouts.
// ---------------------------------------------------------------------------
__global__ __launch_bounds__(256) void emissions_gemm_wmma(
    const float* __restrict__ feat, const float* __restrict__ w,
    const float* __restrict__ bias, float* __restrict__ em)
{
    __shared__ float As[128][34];   // 128 rows x 32 K (pad 2 to dodge bank conflicts)
    __shared__ float Bs[32][32];    // 32 K x 32 N (cols >= 21 zero-padded)

    const int tid  = threadIdx.x;
    const int wave = tid >> 5;
    const int lane = tid & 31;
    const int row0 = blockIdx.x * 128;

    v8f c0 = {0.f, 0.f, 0.f, 0.f, 0.f, 0.f, 0.f, 0.f};
    v8f c1 = {0.f, 0.f, 0.f, 0.f, 0.f, 0.f, 0.f, 0.f};

    // A fragment: lanes 0-15 hold M=lane, K={kk,kk+1}; lanes 16-31 hold M=lane-16, K={kk+2,kk+3}
    const int mrow = wave * 16 + (lane & 15);
    const int kb   = (lane >> 4) << 1;     // 0 or 2
    const int ncol = lane & 15;

    for (int k0 = 0; k0 < ND; k0 += 32) {
        {   // coalesced: each 32-lane group loads one contiguous 128B row slice
            const int r = tid >> 5;
            const int c = tid & 31;
            #pragma unroll
            for (int it = 0; it < 16; ++it)
                As[r + it * 8][c] = feat[(size_t)(row0 + r + it * 8) * ND + k0 + c];
        }
        #pragma unroll
        for (int idx = tid; idx < 32 * 32; idx += 256) {
            const int kk = idx >> 5, nn = idx & 31;
            Bs[kk][nn] = (nn < NT) ? w[(k0 + kk) * NT + nn] : 0.f;
        }
        __syncthreads();

        #pragma unroll
        for (int kk = 0; kk < 32; kk += 4) {
            v2f a, b0, b1;
            a[0]  = As[mrow][kk + kb];        a[1]  = As[mrow][kk + kb + 1];
            b0[0] = Bs[kk + kb][ncol];        b0[1] = Bs[kk + kb + 1][ncol];
            b1[0] = Bs[kk + kb][ncol + 16];   b1[1] = Bs[kk + kb + 1][ncol + 16];
            c0 = __builtin_amdgcn_wmma_f32_16x16x4_f32(false, a, false, b0,
                                                       (short)0, c0, false, false);
            c1 = __builtin_amdgcn_wmma_f32_16x16x4_f32(false, a, false, b1,
                                                       (short)0, c1, false, false);
        }
        __syncthreads();
    }

    // C/D layout: VGPR r -> rows r (lanes 0-15) and r+8 (lanes 16-31), N = lane&15
    const int mbase = wave * 16 + ((lane >> 4) << 3);
    #pragma unroll
    for (int r = 0; r < 8; ++r) {
        const size_t grow = (size_t)(row0 + mbase + r);
        if (ncol < NT)      em[grow * NT + ncol]      = c0[r] + bias[ncol];
        if (ncol + 16 < NT) em[grow * NT + ncol + 16] = c1[r] + bias[ncol + 16];
    }
}

// ---------------------------------------------------------------------------
// Phase 2: CRF forward. One wave32 per batch; lane j owns tag j (j<21).
// Writes per-batch (denominator - numerator) so sum(partials) == -llh.
// ---------------------------------------------------------------------------
__global__ __launch_bounds__(32) void crf_forward_kernel(
    const float* __restrict__ em, const int* __restrict__ labels,
    const int* __restrict__ masks, const float* __restrict__ startT,
    const float* __restrict__ endT, const float* __restrict__ trans,
    float* __restrict__ partials)
{
    const int b = blockIdx.x;
    const int j = threadIdx.x;
    const bool valid = j < NT;
    const int jj = valid ? j : NT - 1;
    const float NEG = -__builtin_inff();

    __shared__ float sT[NT * NT];
    __shared__ float alpha[2][32];

    for (int idx = j; idx < NT * NT; idx += 32) sT[idx] = trans[idx];

    const float* emB = em + (size_t)b * NS * NT;
    const int*   mB  = masks + b * NS;
    const int*   lB  = labels + b * NS;

    alpha[0][j] = valid ? (startT[jj] + emB[jj]) : NEG;
    __syncthreads();

    int cur = 0;
    for (int t = 1; t < NS; ++t) {
        float aj = alpha[cur][j];
        float a_new;
        if (mB[t]) {   // uniform per block
            float m = NEG;
            #pragma unroll
            for (int i = 0; i < NT; ++i)
                m = fmaxf(m, alpha[cur][i] + sT[i * NT + jj]);
            float s = 0.f;
            #pragma unroll
            for (int i = 0; i < NT; ++i)
                s += __expf(alpha[cur][i] + sT[i * NT + jj] - m);
            a_new = m + __logf(s) + emB[t * NT + jj];
        } else {
            a_new = aj;
        }
        alpha[cur ^ 1][j] = valid ? a_new : NEG;
        cur ^= 1;
        __syncthreads();
    }

    // denominator = logsumexp_j(alpha[j] + endT[j])  (wave32 reduce)
    float v = valid ? (alpha[cur][j] + endT[jj]) : NEG;
    float m = v;
    #pragma unroll
    for (int o = 16; o > 0; o >>= 1) m = fmaxf(m, __shfl_xor(m, o, 32));
    float e = __expf(v - m);          // exp(-inf - m) == 0 for invalid lanes
    #pragma unroll
    for (int o = 16; o > 0; o >>= 1) e += __shfl_xor(e, o, 32);
    const float den = m + __logf(e);

    // numerator: distribute t over lanes, also count mask for seq_end
    float part = 0.f;
    int cnt = 0;
    for (int t = j; t < NS; t += 32) {
        const int mk = mB[t];
        cnt += mk ? 1 : 0;
        if (t >= 1 && mk) {
            const int pt = lB[t - 1], ct = lB[t];
            part += sT[pt * NT + ct] + emB[t * NT + ct];
        }
    }
    #pragma unroll
    for (int o = 16; o > 0; o >>= 1) {
        part += __shfl_xor(part, o, 32);
        cnt  += __shfl_xor(cnt, o, 32);
    }

    if (j == 0) {
        const int t0 = lB[0];
        float num = startT[t0] + emB[t0] + part;
        num += endT[lB[cnt - 1]];
        partials[b] = den - num;      // contribution to -llh
    }
}

// ---------------------------------------------------------------------------
// Phase 3: Viterbi decode + in-LDS backtrace. One wave32 per batch.
// ---------------------------------------------------------------------------
__global__ __launch_bounds__(32) void crf_decode_kernel(
    const float* __restrict__ em, const int* __restrict__ masks,
    const float* __restrict__ startT, const float* __restrict__ endT,
    const float* __restrict__ trans, float* __restrict__ out_tags)
{
    const int b = blockIdx.x;
    const int j = threadIdx.x;
    const bool valid = j < NT;
    const int jj = valid ? j : NT - 1;
    const float NEG = -__builtin_inff();

    __shared__ float sT[NT * NT];
    __shared__ float score[2][32];
    __shared__ unsigned char hist[NS - 1][32];

    for (int idx = j; idx < NT * NT; idx += 32) sT[idx] = trans[idx];

    const float* emB = em + (size_t)b * NS * NT;
    const int*   mB  = masks + b * NS;

    score[0][j] = valid ? (startT[jj] + emB[jj]) : NEG;
    __syncthreads();

    int cur = 0;
    for (int t = 1; t < NS; ++t) {
        float sj = score[cur][j];
        float best;
        int bi;
        if (mB[t]) {
            best = NEG; bi = 0;
            #pragma unroll
            for (int i = 0; i < NT; ++i) {
                const float v = score[cur][i] + sT[i * NT + jj];
                if (v > best) { best = v; bi = i; }   // first-max tie-break
            }
            best += emB[t * NT + jj];
        } else {
            best = sj; bi = j;                        // identity when masked
        }
        hist[t - 1][j] = (unsigned char)bi;
        score[cur ^ 1][j] = valid ? best : NEG;
        cur ^= 1;
        __syncthreads();
    }

    // last_tag = argmax_j(score[j] + endT[j]), smallest index on ties
    float v = valid ? (score[cur][j] + endT[jj]) : NEG;
    int idx = j;
    #pragma unroll
    for (int o = 16; o > 0; o >>= 1) {
        const float ov = __shfl_xor(v, o, 32);
        const int   oi = __shfl_xor(idx, o, 32);
        if (ov > v || (ov == v && oi < idx)) { v = ov; idx = oi; }
    }

    if (j == 0) {
        int tag = idx;
        out_tags[(size_t)b * NS + (NS - 1)] = (float)tag;
        for (int t = NS - 2; t >= 0; --t) {
            tag = hist[t][tag];
            out_tags[(size_t)b * NS + t] = (float)tag;
        }
    }
}

// ---------------------------------------------------------------------------
// Phase 4: deterministic tree-reduce of per-batch partials -> d_out[0] = -llh
// ---------------------------------------------------------------------------
__global__ __launch_bounds__(256) void reduce_llh(const float* __restrict__ partials,
                                                  float* __restrict__ out)
{
    __shared__ float s[256];
    const int t = threadIdx.x;
    s[t] = partials[t];
    __syncthreads();
    for (int o = 128; o > 0; o >>= 1) {
        if (t < o) s[t] += s[t + o];
        __syncthreads();
    }
    if (t == 0) out[0] = s[0];
}

extern "C" void kernel_launch(void* const* d_in, const int* in_sizes, int n_in,
                              void* d_out, int out_size, void* d_ws, size_t ws_size,
                              hipStream_t stream)
{
    const float* feat   = (const float*)d_in[0];   // (256,256,768)
    const float* fc_w   = (const float*)d_in[1];   // (768,21)
    const float* fc_b   = (const float*)d_in[2];   // (21,)
    const int*   labels = (const int*)d_in[3];     // (256,256)
    const int*   masks  = (const int*)d_in[4];     // (256,256)
    const float* startT = (const float*)d_in[5];   // (21,)
    const float* endT   = (const float*)d_in[6];   // (21,)
    const float* trans  = (const float*)d_in[7];   // (21,21)

    float* out      = (float*)d_out;               // [ -llh, tags(65536) ]
    float* em       = (float*)d_ws;                // 65536*21 floats (~5.5 MB)
    float* partials = em + (size_t)NROWS * NT;     // 256 floats

    emissions_gemm_wmma<<<NROWS / 128, 256, 0, stream>>>(feat, fc_w, fc_b, em);
    crf_forward_kernel<<<NB, 32, 0, stream>>>(em, labels, masks, startT, endT, trans, partials);
    crf_decode_kernel<<<NB, 32, 0, stream>>>(em, masks, startT, endT, trans, out + 1);
    reduce_llh<<<1, 256, 0, stream>>>(partials, out);
}